// GCN_19567871000962
// MI455X (gfx1250) — compile-verified
//
#include <hip/hip_runtime.h>

// GCN 3-layer forward for MI455X (gfx1250, wave32).
// Aggregation is the bottleneck (L2-resident scatter/gather); GEMMs use
// fp32 WMMA (v_wmma_f32_16x16x4_f32), 16 per output tile.

#define N_NODES 100000
#define N_EDGES 1600000
#define D 64

typedef __attribute__((ext_vector_type(2))) float v2f;
typedef __attribute__((ext_vector_type(8))) float v8f;

__global__ void zero_kernel(float* __restrict__ p, int n) {
  int i = blockIdx.x * blockDim.x + threadIdx.x;
  if (i < n) p[i] = 0.0f;
}

// out-degree: deg[src[e]] += 1  (native global_atomic_add_f32)
__global__ void degree_kernel(const int* __restrict__ src, float* __restrict__ deg) {
  int e = blockIdx.x * blockDim.x + threadIdx.x;
  if (e < N_EDGES) unsafeAtomicAdd(&deg[src[e]], 1.0f);
}

// deg_inv = 1 / max(deg, 1)  (in place)
__global__ void invert_kernel(float* __restrict__ deg) {
  int i = blockIdx.x * blockDim.x + threadIdx.x;
  if (i < N_NODES) deg[i] = 1.0f / fmaxf(deg[i], 1.0f);
}

// hs = h * deg_inv[row], vectorized float4
__global__ void scale_kernel(const float4* __restrict__ h,
                             const float* __restrict__ dinv,
                             float4* __restrict__ hs) {
  int i = blockIdx.x * blockDim.x + threadIdx.x;   // float4 index
  if (i < N_NODES * (D / 4)) {
    int row = i >> 4;                               // 16 float4 per row
    float s = dinv[row];
    float4 v = h[i];
    v.x *= s; v.y *= s; v.z *= s; v.w *= s;
    hs[i] = v;
  }
}

// scatter-add: agg[dst[e], :] += hs[src[e], :]
// 16 lanes per edge, 4 consecutive floats per lane (b128 gather, 4 f32 atomics)
__global__ void scatter_kernel(const int* __restrict__ src,
                               const int* __restrict__ dst,
                               const float* __restrict__ hs,
                               float* __restrict__ agg) {
  long long t = (long long)blockIdx.x * blockDim.x + threadIdx.x;
  int e  = (int)(t >> 4);
  int c4 = (int)(t & 15);
  if (e < N_EDGES) {
    int s = src[e];
    int d = dst[e];
    const float4 v = *(const float4*)(hs + (size_t)s * D + c4 * 4);
    float* base = agg + (size_t)d * D + c4 * 4;
    unsafeAtomicAdd(base + 0, v.x);
    unsafeAtomicAdd(base + 1, v.y);
    unsafeAtomicAdd(base + 2, v.z);
    unsafeAtomicAdd(base + 3, v.w);
  }
}

// out = act(A @ W + bias), A:[N_NODES,64] row-major, W:[64,64] row-major.
// One wave32 per 16x16 tile; K=64 -> 16 x v_wmma_f32_16x16x4_f32.
__global__ __launch_bounds__(256) void gemm_bias_act_kernel(
    const float* __restrict__ A,
    const float* __restrict__ W,
    const float* __restrict__ bias,
    float* __restrict__ out,
    int relu) {
  const int lane = threadIdx.x & 31;
  const int wave = threadIdx.x >> 5;
  const int tile = blockIdx.x * 8 + wave;     // 8 waves per block
  const int tm = tile >> 2;                   // D/16 == 4 N-tiles
  const int tn = tile & 3;
  if (tm >= N_NODES / 16) return;

  const int half = lane >> 4;                 // 0: lanes 0-15, 1: lanes 16-31
  const int l    = lane & 15;
  const int arow = tm * 16 + l;               // A-fragment row (M)
  const int bcol = tn * 16 + l;               // B-fragment / output column (N)

  v8f acc = {};
#pragma unroll
  for (int k0 = 0; k0 < D; k0 += 4) {
    // A 16x4 fragment (ISA 7.12.2): lanes 0-15 hold K=k0,k0+1; lanes 16-31 hold K=k0+2,k0+3
    const float* ap = A + (size_t)arow * D + k0 + 2 * half;
    v2f a; a.x = ap[0]; a.y = ap[1];
    // B 4x16 fragment (mirrored K striping), W[k][n] row-major
    const float* bp = W + (size_t)(k0 + 2 * half) * D + bcol;
    v2f b; b.x = bp[0]; b.y = bp[D];
    acc = __builtin_amdgcn_wmma_f32_16x16x4_f32(
        /*neg_a=*/false, a, /*neg_b=*/false, b,
        /*c_mod=*/(short)0, acc, /*reuse_a=*/false, /*reuse_b=*/false);
  }

  const float bv = bias[bcol];
#pragma unroll
  for (int v = 0; v < 8; ++v) {
    // C/D layout: M = v + 8*half, N = lane&15
    int row = tm * 16 + v + 8 * half;
    float x = acc[v] + bv;
    if (relu) x = fmaxf(x, 0.0f);
    out[(size_t)row * D + bcol] = x;
  }
}

// per-thread column partial sums -> one atomic per thread (4 per column per block)
__global__ void colsum_kernel(const float* __restrict__ h, float* __restrict__ acc) {
  int c  = threadIdx.x & 63;
  int rpb = blockDim.x >> 6;                   // rows handled per block pass (4)
  int r0 = blockIdx.x * rpb + (threadIdx.x >> 6);
  int stride = gridDim.x * rpb;
  float s = 0.0f;
  for (int r = r0; r < N_NODES; r += stride) s += h[(size_t)r * D + c];
  unsafeAtomicAdd(&acc[c], s);
}

__global__ void mean_finalize_kernel(const float* __restrict__ acc,
                                     float* __restrict__ out_mean) {
  int c = threadIdx.x;
  if (c < D) out_mean[c] = acc[c] * (1.0f / (float)N_NODES);
}

extern "C" void kernel_launch(void* const* d_in, const int* in_sizes, int n_in,
                              void* d_out, int out_size, void* d_ws, size_t ws_size,
                              hipStream_t stream) {
  const float* features = (const float*)d_in[0];
  const int*   src      = (const int*)d_in[1];   // harness contract: integer -> const int*
  const int*   dst      = (const int*)d_in[2];
  const float* W0 = (const float*)d_in[3];
  const float* b0 = (const float*)d_in[4];
  const float* W1 = (const float*)d_in[5];
  const float* b1 = (const float*)d_in[6];
  const float* W2 = (const float*)d_in[7];
  const float* b2 = (const float*)d_in[8];
  float* out = (float*)d_out;                    // [N_NODES*D] h, then [D] mean

  // workspace layout (floats): deg_inv | hs | agg | hbuf | mean_acc
  float* ws   = (float*)d_ws;
  float* dinv = ws;
  float* hs   = dinv + N_NODES;
  float* agg  = hs   + (size_t)N_NODES * D;
  float* hbuf = agg  + (size_t)N_NODES * D;
  float* macc = hbuf + (size_t)N_NODES * D;

  const int ND   = N_NODES * D;                  // 6,400,000
  const int B    = 256;
  const int gN   = (N_NODES + B - 1) / B;
  const int gE   = (N_EDGES + B - 1) / B;
  const int gND  = (ND + B - 1) / B;
  const int gND4 = (ND / 4 + B - 1) / B;
  const int gSC  = (int)(((long long)N_EDGES * 16 + B - 1) / B);
  const int gMM  = (N_NODES / 16) * (D / 16) / 8;  // 3125 blocks, 8 tiles each

  // degrees
  zero_kernel<<<gN, B, 0, stream>>>(dinv, N_NODES);
  degree_kernel<<<gE, B, 0, stream>>>(src, dinv);
  invert_kernel<<<gN, B, 0, stream>>>(dinv);

  // layer 0: features -> hbuf (ReLU)
  scale_kernel<<<gND4, B, 0, stream>>>((const float4*)features, dinv, (float4*)hs);
  zero_kernel<<<gND, B, 0, stream>>>(agg, ND);
  scatter_kernel<<<gSC, B, 0, stream>>>(src, dst, hs, agg);
  gemm_bias_act_kernel<<<gMM, B, 0, stream>>>(agg, W0, b0, hbuf, 1);

  // layer 1: hbuf -> hbuf (ReLU; GEMM reads only agg, so in/out alias is safe)
  scale_kernel<<<gND4, B, 0, stream>>>((const float4*)hbuf, dinv, (float4*)hs);
  zero_kernel<<<gND, B, 0, stream>>>(agg, ND);
  scatter_kernel<<<gSC, B, 0, stream>>>(src, dst, hs, agg);
  gemm_bias_act_kernel<<<gMM, B, 0, stream>>>(agg, W1, b1, hbuf, 1);

  // layer 2: hbuf -> d_out (no act)
  scale_kernel<<<gND4, B, 0, stream>>>((const float4*)hbuf, dinv, (float4*)hs);
  zero_kernel<<<gND, B, 0, stream>>>(agg, ND);
  scatter_kernel<<<gSC, B, 0, stream>>>(src, dst, hs, agg);
  gemm_bias_act_kernel<<<gMM, B, 0, stream>>>(agg, W2, b2, out, 0);

  // mean readout
  zero_kernel<<<1, 64, 0, stream>>>(macc, 64);
  colsum_kernel<<<512, B, 0, stream>>>(out, macc);
  mean_finalize_kernel<<<1, 64, 0, stream>>>(macc, out + (size_t)N_NODES * D);
}